// SimplifiedHEPT_75196287418966
// MI455X (gfx1250) — compile-verified
//
#include <hip/hip_runtime.h>

// CDNA5 / gfx1250: wave32, WMMA 16x16 shapes.
// out[h,n,m] = exp(-0.5 * max(0, |q_n|^2 + |k_m|^2 - 2 q_n.k_m))
// Strategy: 128x128 output tile per block, fp32 WMMA (16x16x4) GEMM for q.k,
// row norms computed once per tile into LDS, fused clamp+exp epilogue.

typedef float v2f __attribute__((ext_vector_type(2)));
typedef float v8f __attribute__((ext_vector_type(8)));

#define N_CTX 4096
#define HEADS 8
#define DIM   64
#define TILE  128
// 64 data floats + 4 pad: rows stay 16B aligned for float4 fills, and the
// per-fragment ds_load_b64 pattern (banks (4*M + c) mod 64) is conflict-free.
#define LDSW  68

__global__ __launch_bounds__(256)
void hept_rbf_wmma_kernel(const float* __restrict__ Q,
                          const float* __restrict__ K,
                          float* __restrict__ Out) {
  __shared__ __attribute__((aligned(16))) float Qs[TILE * LDSW];
  __shared__ __attribute__((aligned(16))) float Ks[TILE * LDSW];
  __shared__ float q2s[TILE];
  __shared__ float k2s[TILE];

  const int h       = blockIdx.z;
  const int rowBase = blockIdx.y * TILE;   // query rows of this tile
  const int colBase = blockIdx.x * TILE;   // key rows (output cols)
  const int tid     = threadIdx.x;

  const float* __restrict__ Qh = Q + (size_t)h * N_CTX * DIM;
  const float* __restrict__ Kh = K + (size_t)h * N_CTX * DIM;

  // ---- cooperative tile load: 128 rows x 64 floats = 2048 float4 per tensor,
  //      256 threads x 8 float4 each (fully coalesced b128 loads) ----
#pragma unroll
  for (int i = 0; i < 8; ++i) {
    int idx = tid + i * 256;        // 0..2047
    int r   = idx >> 4;             // row 0..127
    int c4  = (idx & 15) << 2;      // col 0,4,...,60
    float4 qv = *(const float4*)(Qh + (size_t)(rowBase + r) * DIM + c4);
    float4 kv = *(const float4*)(Kh + (size_t)(colBase + r) * DIM + c4);
    *(float4*)(Qs + r * LDSW + c4) = qv;
    *(float4*)(Ks + r * LDSW + c4) = kv;
  }
  __syncthreads();

  // ---- row norms: threads 0..127 -> |q|^2, threads 128..255 -> |k|^2 ----
  {
    const float* row = (tid < TILE) ? (Qs + tid * LDSW)
                                    : (Ks + (tid - TILE) * LDSW);
    float s = 0.f;
#pragma unroll
    for (int c = 0; c < DIM; ++c) s = fmaf(row[c], row[c], s);
    if (tid < TILE) q2s[tid] = s; else k2s[tid - TILE] = s;
  }
  __syncthreads();

  // ---- wave-level WMMA GEMM: each wave owns a 32x64 subtile (2x4 fragments) ----
  const int wave    = tid >> 5;            // 0..7 (wave32)
  const int lane    = tid & 31;
  const int lane16  = lane & 15;
  const int hi      = lane >> 4;           // 0: K=0,1 / row M   1: K=2,3 / row M+8
  const int waveRow = (wave >> 1) * 32;    // 0,32,64,96
  const int waveCol = (wave & 1) * 64;     // 0,64
  const int koff    = hi * 2;              // lane-half K offset for A/B fragments

  v8f acc[2][4];
#pragma unroll
  for (int i = 0; i < 2; ++i)
#pragma unroll
    for (int j = 0; j < 4; ++j)
#pragma unroll
      for (int r = 0; r < 8; ++r) acc[i][j][r] = 0.0f;

#pragma unroll
  for (int k0 = 0; k0 < DIM; k0 += 4) {
    // A fragment (16x4 f32): lane L -> row M=L%16, two consecutive K values.
    v2f a[2];
#pragma unroll
    for (int i = 0; i < 2; ++i)
      a[i] = *(const v2f*)(Qs + (waveRow + i * 16 + lane16) * LDSW + k0 + koff);
    // B fragment (4x16 f32) = K^T tile: lane L -> col N=L%16 (= key row),
    // two consecutive d values -> single ds_load_b64 from row-major K tile.
    v2f b[4];
#pragma unroll
    for (int j = 0; j < 4; ++j)
      b[j] = *(const v2f*)(Ks + (waveCol + j * 16 + lane16) * LDSW + k0 + koff);

#pragma unroll
    for (int i = 0; i < 2; ++i)
#pragma unroll
      for (int j = 0; j < 4; ++j)
        acc[i][j] = __builtin_amdgcn_wmma_f32_16x16x4_f32(
            /*neg_a=*/false, a[i], /*neg_b=*/false, b[j],
            /*c_mod=*/(short)0, acc[i][j],
            /*reuse_a=*/false, /*reuse_b=*/false);
  }

  // ---- fused epilogue: d = q2 + k2 - 2*qk, clamp, exp(-0.5 d), store ----
  // C/D layout: VGPR r -> row (r + 8*hi), col = lane16.
  float* __restrict__ OutH = Out + (size_t)h * N_CTX * N_CTX;
#pragma unroll
  for (int i = 0; i < 2; ++i) {
    const int rLoc0 = waveRow + i * 16 + hi * 8;
#pragma unroll
    for (int j = 0; j < 4; ++j) {
      const int cLoc = waveCol + j * 16 + lane16;
      const float k2v = k2s[cLoc];
      float* __restrict__ outCol = OutH + (size_t)(rowBase + rLoc0) * N_CTX
                                        + colBase + cLoc;
#pragma unroll
      for (int r = 0; r < 8; ++r) {
        float d = q2s[rLoc0 + r] + k2v - 2.0f * acc[i][j][r];
        d = fmaxf(d, 0.0f);
        outCol[(size_t)r * N_CTX] = __expf(-0.5f * d);
      }
    }
  }
}

extern "C" void kernel_launch(void* const* d_in, const int* in_sizes, int n_in,
                              void* d_out, int out_size, void* d_ws, size_t ws_size,
                              hipStream_t stream) {
  (void)in_sizes; (void)n_in; (void)out_size; (void)d_ws; (void)ws_size;
  const float* Q = (const float*)d_in[0];
  const float* K = (const float*)d_in[1];
  float* Out     = (float*)d_out;
  dim3 grid(N_CTX / TILE, N_CTX / TILE, HEADS);  // 32 x 32 x 8 blocks
  hept_rbf_wmma_kernel<<<grid, 256, 0, stream>>>(Q, K, Out);
}